// GARPNHead_23244363006288
// MI455X (gfx1250) — compile-verified
//
#include <hip/hip_runtime.h>

// ---------------------------------------------------------------------------
// GARPN head for MI455X (gfx1250, wave32).
// Heavy convs run as implicit-im2col GEMMs on V_WMMA_F32_16X16X4_F32
// (fp32 matrix pipe: reference is fp32, problem is compute-bound).
// ---------------------------------------------------------------------------

typedef float v2f __attribute__((ext_vector_type(2)));
typedef float v8f __attribute__((ext_vector_type(8)));

#define Bn 2
#define Cn 256
#define Hn 96
#define Wn 96
#define HWn (Hn * Wn)          // 9216
#define DGn 4
#define CGn 64                 // Cn / DGn
#define KKn 9
#define WTILES (Wn / 16)       // 6
// swizzled weight fragment tensor: [16 ot][9 kk][64 kq][32 lane][2]
#define WT_ELEMS (16 * 9 * 64 * 32 * 2)   // 589824

__device__ __forceinline__ v8f wmma_f32(v2f a, v2f b, v8f c) {
  // (neg_a, A, neg_b, B, c_mod, C, reuse_a, reuse_b)
  return __builtin_amdgcn_wmma_f32_16x16x4_f32(false, a, false, b, (short)0, c,
                                               false, false);
}

// ---------------------------------------------------------------------------
// Swizzle [Co,Ci,3,3] weights into per-lane WMMA A-fragment order so the main
// loops do fully-coalesced 8B/lane loads: wT[ot][kk][kq][lane][2], where
// M = lane%16 (out ch within 16-tile), K-half selected by lane>=16.
// ---------------------------------------------------------------------------
__global__ void transform_weights(const float* __restrict__ w,
                                  float* __restrict__ wT) {
  int idx = blockIdx.x * blockDim.x + threadIdx.x;
  if (idx >= WT_ELEMS) return;
  int j    = idx & 1;
  int lane = (idx >> 1) & 31;
  int kq   = (idx >> 6) & 63;
  int okk  = idx >> 12;          // ot*9 + kk
  int kk   = okk % 9;
  int ot   = okk / 9;
  int m    = lane & 15;
  int oc   = ot * 16 + m;
  int cin  = kq * 4 + ((lane >= 16) ? 2 : 0) + j;
  wT[idx] = w[(oc * Cn + cin) * 9 + kk];
}

// ---------------------------------------------------------------------------
// 3x3 conv (pad 1) + bias + ReLU via WMMA. Block = 256 thr (8 waves) owns one
// (b, h, 16-pixel) tile and all 256 out channels. LDS stages 64ch x 16px.
// ---------------------------------------------------------------------------
__global__ __launch_bounds__(256) void conv3x3_relu_wmma(
    const float* __restrict__ x, const float* __restrict__ wT,
    const float* __restrict__ bias, float* __restrict__ out) {
  __shared__ float lds[64 * 16];

  int blk = blockIdx.x;
  int b   = blk / (Hn * WTILES);
  int rem = blk % (Hn * WTILES);
  int h   = rem / WTILES;
  int w0  = (rem % WTILES) * 16;

  int tid   = threadIdx.x;
  int lane  = tid & 31;
  int wv    = tid >> 5;          // 0..7
  int hi    = lane >> 4;         // 0 or 1
  int n     = lane & 15;         // pixel column in tile (B/D matrix N)
  int khalf = hi * 2;            // B-fragment K half
  int mhalf = hi * 8;            // D-matrix M half
  int ot0   = wv * 2;            // two 16-ch out tiles per wave

  v8f acc[2];
#pragma unroll
  for (int t2 = 0; t2 < 2; ++t2) {
    int otb = (ot0 + t2) * 16;
#pragma unroll
    for (int r = 0; r < 8; ++r) acc[t2][r] = bias[otb + r + mhalf];
  }

  const int ci0 = tid >> 4;      // 0..15 (staging channel)
  const int wi  = tid & 15;      // 0..15 (staging pixel)

  for (int kk = 0; kk < 9; ++kk) {
    int ky = kk / 3, kx = kk % 3;
    int yy = h + ky - 1;
    int xx = w0 + wi + kx - 1;
    bool inb = ((unsigned)yy < (unsigned)Hn) && ((unsigned)xx < (unsigned)Wn);
    for (int cc = 0; cc < 4; ++cc) {
      __syncthreads();
#pragma unroll
      for (int s = 0; s < 4; ++s) {
        int ci = ci0 + s * 16;
        int c  = cc * 64 + ci;
        float v = 0.f;
        if (inb) v = x[((b * Cn + c) * Hn + yy) * Wn + xx];
        lds[ci * 16 + wi] = v;
      }
      __syncthreads();
      for (int kq = 0; kq < 16; ++kq) {
        int k0 = kq * 4;
        v2f bfrag;
        bfrag.x = lds[(k0 + khalf) * 16 + n];
        bfrag.y = lds[(k0 + khalf + 1) * 16 + n];
        int kqf = cc * 16 + kq;
#pragma unroll
        for (int t2 = 0; t2 < 2; ++t2) {
          const float* ap =
              wT + (((((ot0 + t2) * 9 + kk) * 64 + kqf) * 32 + lane) * 2);
          v2f afrag;
          afrag.x = ap[0];
          afrag.y = ap[1];
          acc[t2] = wmma_f32(afrag, bfrag, acc[t2]);
        }
      }
    }
  }

#pragma unroll
  for (int t2 = 0; t2 < 2; ++t2) {
    int otb = (ot0 + t2) * 16;
#pragma unroll
    for (int r = 0; r < 8; ++r) {
      int oc  = otb + r + mhalf;
      float v = acc[t2][r];
      v = v > 0.f ? v : 0.f;
      out[((b * Cn + oc) * Hn + h) * Wn + w0 + n] = v;
    }
  }
}

// ---------------------------------------------------------------------------
// Deformable 3x3 conv (4 groups) + ReLU via WMMA. Same GEMM skeleton; the LDS
// staging does the bilinear gather from t using per-(g,kk,pixel) offsets.
// ---------------------------------------------------------------------------
__global__ __launch_bounds__(256) void deform_conv_relu_wmma(
    const float* __restrict__ t, const float* __restrict__ offs,
    const float* __restrict__ wT, float* __restrict__ out) {
  __shared__ float lds[64 * 16];

  int blk = blockIdx.x;
  int b   = blk / (Hn * WTILES);
  int rem = blk % (Hn * WTILES);
  int h   = rem / WTILES;
  int w0  = (rem % WTILES) * 16;

  int tid   = threadIdx.x;
  int lane  = tid & 31;
  int wv    = tid >> 5;
  int hi    = lane >> 4;
  int n     = lane & 15;
  int khalf = hi * 2;
  int mhalf = hi * 8;
  int ot0   = wv * 2;

  v8f acc[2];
#pragma unroll
  for (int t2 = 0; t2 < 2; ++t2)
#pragma unroll
    for (int r = 0; r < 8; ++r) acc[t2][r] = 0.f;

  const int ci0 = tid >> 4;
  const int wi  = tid & 15;
  const int pix = h * Wn + w0 + wi;

  for (int g = 0; g < DGn; ++g) {
    for (int kk = 0; kk < 9; ++kk) {
      int ky = kk / 3, kx = kk % 3;
      float dy = offs[(b * 72 + (g * 18 + kk * 2 + 0)) * HWn + pix];
      float dx = offs[(b * 72 + (g * 18 + kk * 2 + 1)) * HWn + pix];
      float py = (float)(h + ky - 1) + dy;
      float px = (float)(w0 + wi + kx - 1) + dx;
      float y0f = floorf(py), x0f = floorf(px);
      float wy = py - y0f, wx = px - x0f;
      int y0 = (int)y0f, x0 = (int)x0f;
      float w00 = (1.f - wy) * (1.f - wx);
      float w01 = (1.f - wy) * wx;
      float w10 = wy * (1.f - wx);
      float w11 = wy * wx;
      bool vy0 = (unsigned)y0 < (unsigned)Hn;
      bool vy1 = (unsigned)(y0 + 1) < (unsigned)Hn;
      bool vx0 = (unsigned)x0 < (unsigned)Wn;
      bool vx1 = (unsigned)(x0 + 1) < (unsigned)Wn;
      int y0c = min(max(y0, 0), Hn - 1);
      int y1c = min(max(y0 + 1, 0), Hn - 1);
      int x0c = min(max(x0, 0), Wn - 1);
      int x1c = min(max(x0 + 1, 0), Wn - 1);

      __syncthreads();
#pragma unroll
      for (int s = 0; s < 4; ++s) {
        int ci = ci0 + s * 16;
        int c  = g * CGn + ci;
        const float* base = t + (b * Cn + c) * HWn;
        float v = 0.f;
        if (vy0) {
          const float* row = base + y0c * Wn;
          if (vx0) v += w00 * row[x0c];
          if (vx1) v += w01 * row[x1c];
        }
        if (vy1) {
          const float* row = base + y1c * Wn;
          if (vx0) v += w10 * row[x0c];
          if (vx1) v += w11 * row[x1c];
        }
        lds[ci * 16 + wi] = v;
      }
      __syncthreads();
      for (int kq = 0; kq < 16; ++kq) {
        int k0 = kq * 4;
        v2f bfrag;
        bfrag.x = lds[(k0 + khalf) * 16 + n];
        bfrag.y = lds[(k0 + khalf + 1) * 16 + n];
        int kqf = g * 16 + kq;
#pragma unroll
        for (int t2 = 0; t2 < 2; ++t2) {
          const float* ap =
              wT + (((((ot0 + t2) * 9 + kk) * 64 + kqf) * 32 + lane) * 2);
          v2f afrag;
          afrag.x = ap[0];
          afrag.y = ap[1];
          acc[t2] = wmma_f32(afrag, bfrag, acc[t2]);
        }
      }
    }
  }

#pragma unroll
  for (int t2 = 0; t2 < 2; ++t2) {
    int otb = (ot0 + t2) * 16;
#pragma unroll
    for (int r = 0; r < 8; ++r) {
      int oc  = otb + r + mhalf;
      float v = acc[t2][r];
      v = v > 0.f ? v : 0.f;
      out[((b * Cn + oc) * Hn + h) * Wn + w0 + n] = v;
    }
  }
}

// ---------------------------------------------------------------------------
// Fused 1x1 heads on t: loc (1ch), shape_pred (2ch), and the 72-channel
// offset projection of shape_pred. One thread per (b, pixel).
// ---------------------------------------------------------------------------
__global__ __launch_bounds__(256) void heads1(
    const float* __restrict__ t, const float* __restrict__ loc_w,
    const float* __restrict__ loc_b, const float* __restrict__ shape_w,
    const float* __restrict__ shape_b, const float* __restrict__ offset_w,
    float* __restrict__ out_loc, float* __restrict__ out_shape,
    float* __restrict__ offs) {
  int idx = blockIdx.x * blockDim.x + threadIdx.x;
  if (idx >= Bn * HWn) return;
  int b   = idx / HWn;
  int pix = idx % HWn;
  const float* tb = t + (size_t)b * Cn * HWn + pix;
  float al = 0.f, a0 = 0.f, a1 = 0.f;
  for (int c = 0; c < Cn; ++c) {
    float v = tb[c * HWn];
    al += v * loc_w[c];
    a0 += v * shape_w[c];
    a1 += v * shape_w[Cn + c];
  }
  out_loc[idx] = al + loc_b[0];
  float s0 = a0 + shape_b[0];
  float s1 = a1 + shape_b[1];
  out_shape[(b * 2 + 0) * HWn + pix] = s0;
  out_shape[(b * 2 + 1) * HWn + pix] = s1;
  for (int o = 0; o < 72; ++o)
    offs[(b * 72 + o) * HWn + pix] = s0 * offset_w[o * 2] + s1 * offset_w[o * 2 + 1];
}

// ---------------------------------------------------------------------------
// Fused 1x1 heads on ta: cls logits (1ch) + bbox (4ch).
// ---------------------------------------------------------------------------
__global__ __launch_bounds__(256) void heads2(
    const float* __restrict__ ta, const float* __restrict__ cls_w,
    const float* __restrict__ cls_b, const float* __restrict__ bbox_w,
    const float* __restrict__ bbox_b, float* __restrict__ out_logits,
    float* __restrict__ out_bbox) {
  int idx = blockIdx.x * blockDim.x + threadIdx.x;
  if (idx >= Bn * HWn) return;
  int b   = idx / HWn;
  int pix = idx % HWn;
  const float* tb = ta + (size_t)b * Cn * HWn + pix;
  float ac = 0.f, ab0 = 0.f, ab1 = 0.f, ab2 = 0.f, ab3 = 0.f;
  for (int c = 0; c < Cn; ++c) {
    float v = tb[c * HWn];
    ac  += v * cls_w[c];
    ab0 += v * bbox_w[0 * Cn + c];
    ab1 += v * bbox_w[1 * Cn + c];
    ab2 += v * bbox_w[2 * Cn + c];
    ab3 += v * bbox_w[3 * Cn + c];
  }
  out_logits[idx] = ac + cls_b[0];
  out_bbox[(b * 4 + 0) * HWn + pix] = ab0 + bbox_b[0];
  out_bbox[(b * 4 + 1) * HWn + pix] = ab1 + bbox_b[1];
  out_bbox[(b * 4 + 2) * HWn + pix] = ab2 + bbox_b[2];
  out_bbox[(b * 4 + 3) * HWn + pix] = ab3 + bbox_b[3];
}

// ---------------------------------------------------------------------------
extern "C" void kernel_launch(void* const* d_in, const int* in_sizes, int n_in,
                              void* d_out, int out_size, void* d_ws,
                              size_t ws_size, hipStream_t stream) {
  const float* feature  = (const float*)d_in[0];
  const float* conv_w   = (const float*)d_in[1];
  const float* conv_b   = (const float*)d_in[2];
  const float* loc_w    = (const float*)d_in[3];
  const float* loc_b    = (const float*)d_in[4];
  const float* shape_w  = (const float*)d_in[5];
  const float* shape_b  = (const float*)d_in[6];
  const float* offset_w = (const float*)d_in[7];
  const float* adapt_w  = (const float*)d_in[8];
  const float* cls_w    = (const float*)d_in[9];
  const float* cls_b    = (const float*)d_in[10];
  const float* bbox_w   = (const float*)d_in[11];
  const float* bbox_b   = (const float*)d_in[12];

  float* ws    = (float*)d_ws;
  float* wT1   = ws;                         // 589824
  float* wT2   = wT1 + WT_ELEMS;             // 589824
  float* tbuf  = wT2 + WT_ELEMS;             // 2*256*9216
  float* tabuf = tbuf + (size_t)Bn * Cn * HWn;
  float* offbf = tabuf + (size_t)Bn * Cn * HWn;  // 2*72*9216

  float* out        = (float*)d_out;
  float* out_logits = out;                          // [2,1,96,96]
  float* out_bbox   = out + Bn * 1 * HWn;           // [2,4,96,96]
  float* out_shape  = out + Bn * 5 * HWn;           // [2,2,96,96]
  float* out_loc    = out + Bn * 7 * HWn;           // [2,1,96,96]

  const int tw_blocks = (WT_ELEMS + 255) / 256;
  transform_weights<<<tw_blocks, 256, 0, stream>>>(conv_w, wT1);
  transform_weights<<<tw_blocks, 256, 0, stream>>>(adapt_w, wT2);

  const int conv_blocks = Bn * Hn * WTILES;  // 1152
  conv3x3_relu_wmma<<<conv_blocks, 256, 0, stream>>>(feature, wT1, conv_b, tbuf);

  const int px_blocks = (Bn * HWn + 255) / 256;  // 72
  heads1<<<px_blocks, 256, 0, stream>>>(tbuf, loc_w, loc_b, shape_w, shape_b,
                                        offset_w, out_loc, out_shape, offbf);

  deform_conv_relu_wmma<<<conv_blocks, 256, 0, stream>>>(tbuf, offbf, wT2,
                                                         tabuf);

  heads2<<<px_blocks, 256, 0, stream>>>(tabuf, cls_w, cls_b, bbox_w, bbox_b,
                                        out_logits, out_bbox);
}